// SpikingConv2D_16904991277738
// MI455X (gfx1250) — compile-verified
//
#include <hip/hip_runtime.h>

typedef __attribute__((ext_vector_type(2))) float v2f;
typedef __attribute__((ext_vector_type(8))) float v8f;

#define HH 256
#define WW 256
#define NF 64
#define HROWS 4

// Branch-free zero-padded pixel load: clamp the address (always legal), then
// select 0 when logically out of bounds. Avoids exec-mask save/restore
// diamonds around every VMEM op.
__device__ __forceinline__ float ldpx(const float* __restrict__ inb, int hh, int ww) {
    const bool ok = ((unsigned)hh < (unsigned)HH) && ((unsigned)ww < (unsigned)WW);
    int hc = hh; if (hc < 0) hc = 0; if (hc > HH - 1) hc = HH - 1;
    int wc = ww; if (wc < 0) wc = 0; if (wc > WW - 1) wc = WW - 1;
    const float v = inb[hc * WW + wc];
    return ok ? v : 0.0f;
}

// One wave computes 16(filter) x 16(pixel) fp32 output tiles of the 3x3 conv
// as D = A(16x12) * B(12x16) via three chained V_WMMA_F32_16X16X4_F32 per
// output row, iterating HROWS rows with a rolling 3-row input window.
__global__ __launch_bounds__(256) void spiking_conv2d_wmma(
    const float* __restrict__ tj,   // (32,1,256,256)
    const float* __restrict__ wt,   // (64,1,3,3)
    const float* __restrict__ bb,   // (64,1)
    const float* __restrict__ di,   // (9,64), row 0 used
    float* __restrict__ out)        // (32,64,256,256)
{
    const int lane = threadIdx.x & 31;
    const int wave = threadIdx.x >> 5;
    const unsigned gw = blockIdx.x * 8u + wave;

    // gw bits: [1:0]=ftile(4) [5:2]=wtile(16) [11:6]=hblk(64) [16:12]=n(32)
    const int ftile = gw & 3;
    const int wtile = (gw >> 2) & 15;
    const int hblk  = (gw >> 6) & 63;
    const int n     = gw >> 12;

    const int fbase = ftile << 4;
    const int wbase = wtile << 4;
    const int hbase = hblk << 2;
    const bool hsel = (lane & 16) != 0;  // which K pair this lane holds
    const int  half = hsel ? 1 : 0;
    const int  l15  = lane & 15;         // M (for A) / N (for B,C,D)

    // ---- A operands: weights, K padded 9 -> 12 with zeros -----------------
    // 32-bit A 16x4 layout: VGPR0 = K{0,2}, VGPR1 = K{1,3} split by lane half.
    v2f a[3];
    {
        const int f = fbase + l15;
#pragma unroll
        for (int ks = 0; ks < 3; ++ks) {
            const int k0 = ks * 4 + half * 2;
            a[ks].x = (k0     < 9) ? wt[f * 9 + k0    ] : 0.0f;
            a[ks].y = (k0 + 1 < 9) ? wt[f * 9 + k0 + 1] : 0.0f;
        }
    }

    // ---- per-lane output constants: c[f] = 1 - D_i[0,f] - B[f] ------------
    float cadd[8];
#pragma unroll
    for (int r = 0; r < 8; ++r) {
        const int f = fbase + half * 8 + r;
        cadd[r] = 1.0f - di[f] - bb[f];
    }

    const float* inb = tj + (size_t)n * HH * WW;
    const int wcol = wbase + l15;

    // Rolling 3-row window: rX[c] = in[hh][wcol + c - 1], zero padded.
    float r0[3], r1[3], r2[3];
#pragma unroll
    for (int c = 0; c < 3; ++c) {
        r0[c] = ldpx(inb, hbase - 1, wcol + c - 1);
        r1[c] = ldpx(inb, hbase    , wcol + c - 1);
    }

#pragma unroll
    for (int hr = 0; hr < HROWS; ++hr) {
        const int h = hbase + hr;
#pragma unroll
        for (int c = 0; c < 3; ++c)
            r2[c] = ldpx(inb, h + 1, wcol + c - 1);

        // B operand selection (all indices compile-time; one cndmask each):
        //   b[ks].x = K = 4ks + 2*half ; b[ks].y = K = 4ks + 1 + 2*half
        //   K -> X[k] with X[0..2]=r0, X[3..5]=r1, X[6..8]=r2, X[>=9]=0
        v2f b0, b1, b2;
        b0.x = hsel ? r0[2] : r0[0];
        b0.y = hsel ? r1[0] : r0[1];
        b1.x = hsel ? r2[0] : r1[1];
        b1.y = hsel ? r2[1] : r1[2];
        b2.x = hsel ? 0.0f  : r2[2];
        b2.y = 0.0f;

        v8f acc = {};
        acc = __builtin_amdgcn_wmma_f32_16x16x4_f32(
            false, a[0], false, b0, (short)0, acc, false, false);
        acc = __builtin_amdgcn_wmma_f32_16x16x4_f32(
            false, a[1], false, b1, (short)0, acc, false, false);
        acc = __builtin_amdgcn_wmma_f32_16x16x4_f32(
            false, a[2], false, b2, (short)0, acc, false, false);

        // ---- epilogue: add constant, clamp, store -------------------------
        float* op = out + (((size_t)n * NF + fbase + half * 8) * HH + h) * WW + wcol;
#pragma unroll
        for (int r = 0; r < 8; ++r) {
            float v = acc[r] + cadd[r];
            v = fminf(v, 1.0f);
            op[(size_t)r * HH * WW] = v;
        }

        // shift the window
#pragma unroll
        for (int c = 0; c < 3; ++c) { r0[c] = r1[c]; r1[c] = r2[c]; }
    }
}

extern "C" void kernel_launch(void* const* d_in, const int* in_sizes, int n_in,
                              void* d_out, int out_size, void* d_ws, size_t ws_size,
                              hipStream_t stream) {
    const float* tj = (const float*)d_in[0];  // (32,1,256,256) fp32
    const float* wt = (const float*)d_in[1];  // (64,1,3,3)     fp32
    const float* bb = (const float*)d_in[2];  // (64,1)         fp32
    const float* di = (const float*)d_in[3];  // (9,64)         fp32
    float* out = (float*)d_out;               // (32,64,256,256) fp32

    // waves = 32 (n) * 64 (hblk) * 16 (wtile) * 4 (ftile) = 131072
    // blocks = 131072 / 8 waves per 256-thread block = 16384
    dim3 grid(16384), block(256);
    spiking_conv2d_wmma<<<grid, block, 0, stream>>>(tj, wt, bb, di, out);
}